// RetinaNetDecoder_15839839388224
// MI455X (gfx1250) — compile-verified
//
#include <hip/hip_runtime.h>
#include <hip/hip_bf16.h>
#include <math.h>

// RetinaNet decode + greedy class-aware NMS for MI455X (gfx1250).
//
// Phase 1 (streaming, HBM-bound ~9.5us): per-anchor class max/argmax + box
//   decode. cls_heads tiles (128 anchors x 80 classes x 4B = 40KB) are DMA'd
//   into LDS with the CDNA5 Tensor Data Mover (tensor_load_to_lds +
//   s_wait_tensorcnt), then reduced from LDS with a staggered, bank-conflict-
//   free access pattern.
// Phase 2 (latency-bound serial NMS): one workgroup per batch image. The live
//   score vector (76725 floats = 300KB) fits in the 320KB per-WGP LDS, so all
//   50 argmax+suppress sweeps run out of LDS; boxes/areas/classes are L2-
//   resident read-only streams. wave32 shfl reductions (warpSize==32).

#define BATCH    8
#define NANCH    76725
#define NCLS     80
#define MAX_DET  50
#define SCORE_THR 0.1f
#define IOU_THR   0.5f
#define IMG_WM1   639.0f
#define IMG_HM1   639.0f

#define TILE       128     // anchors per phase-1 workgroup (LDS tile = 40KB)
#define P2_THREADS 1024    // 32 waves, one workgroup per batch image

typedef __attribute__((ext_vector_type(4))) unsigned int tdm_u32x4;
typedef __attribute__((ext_vector_type(8))) int          tdm_i32x8;
typedef __attribute__((ext_vector_type(4))) int          tdm_i32x4;

// ---------------------------------------------------------------------------
// Phase 1: class max/argmax + box decode.
// ---------------------------------------------------------------------------
__global__ __launch_bounds__(TILE) void retina_decode_kernel(
    const float* __restrict__ cls, const float* __restrict__ reg,
    const float* __restrict__ anc,
    float* __restrict__ boxes, float* __restrict__ areas,
    float* __restrict__ scores, int* __restrict__ classes)
{
  // Must be the ONLY static LDS object in this kernel: TDM descriptor uses
  // lds_addr == 0 (start of the workgroup's LDS allocation).
  __shared__ float tile[TILE * NCLS];

  const int b  = blockIdx.y;
  const int t0 = blockIdx.x * TILE;
  const int t  = threadIdx.x;
  const int a  = t0 + t;
  const long long rowbase = (long long)b * NANCH + t0;

  float best = -1e30f;
  int   bcls = 0;

#if __has_builtin(__builtin_amdgcn_tensor_load_to_lds)
  {
    int rows = NANCH - t0;
    if (rows > TILE) rows = TILE;
    if (t < 32) {  // one wave issues the TDM descriptor (EXEC-independent op)
      unsigned long long gaddr =
          (unsigned long long)(const void*)cls +
          (unsigned long long)rowbase * (NCLS * 4ull);
      // D# group 0: count=1 (valid), is_restore=0, gather off, lds_addr=0,
      // global_addr[56:0], type=2 ("image").
      tdm_u32x4 g0;
      g0[0] = 1u;
      g0[1] = 0u;                                        // lds_addr
      g0[2] = (unsigned)(gaddr & 0xFFFFFFFFull);         // global_addr[31:0]
      g0[3] = (unsigned)((gaddr >> 32) & 0x1FFFFFFull)   // global_addr[56:32]
              | (2u << 30);                              // type=2
      // D# group 1: data_size=2 (4B), tensor_dim0=80, tensor_dim1=rows,
      // tile_dim0=80, tile_dim1=TILE (OOB rows return zero), dim0_stride=80.
      tdm_i32x8 g1;
      g1[0] = (2 << 16);                                 // data_size = 4B
      g1[1] = (NCLS & 0xFFFF) << 16;                     // tensor_dim0 lo16
      g1[2] = (rows & 0xFFFF) << 16;                     // dim0 hi16=0 | dim1 lo16
      g1[3] = ((rows >> 16) & 0xFFFF) | (NCLS << 16);    // dim1 hi16 | tile_dim0
      g1[4] = TILE;                                      // tile_dim1 | tile_dim2=0
      g1[5] = NCLS;                                      // tensor_dim0_stride lo32
      g1[6] = 0;                                         // stride hi16 | dim1_stride lo16
      g1[7] = 0;
      tdm_i32x4 gz4; gz4[0] = 0; gz4[1] = 0; gz4[2] = 0; gz4[3] = 0;
      tdm_i32x8 gz8;
      gz8[0] = 0; gz8[1] = 0; gz8[2] = 0; gz8[3] = 0;
      gz8[4] = 0; gz8[5] = 0; gz8[6] = 0; gz8[7] = 0;
      // clang-23 / therock toolchain: 6-arg form (extra int32x8 group).
      __builtin_amdgcn_tensor_load_to_lds(g0, g1, gz4, gz4, gz8, 0);
      __builtin_amdgcn_s_wait_tensorcnt(0);
    }
    __syncthreads();
    // Staggered start (effective stride 80+5=85 dwords, odd vs 64 banks ->
    // conflict-free across a wave). Tie rule is order-independent and matches
    // jnp.argmax (first/lowest class index wins on equal scores).
    const int base = t * NCLS;
    int k0 = (t * 5) % NCLS;
    for (int kk = 0; kk < NCLS; ++kk) {
      int k = k0 + kk; if (k >= NCLS) k -= NCLS;
      float v = tile[base + k];
      if (v > best || (v == best && k < bcls)) { best = v; bcls = k; }
    }
  }
#else
  if (a < NANCH) {
    const float4* cp =
        reinterpret_cast<const float4*>(cls + (long long)(rowbase + t) * NCLS);
    for (int j = 0; j < NCLS / 4; ++j) {
      float4 q = cp[j];
      int k = j * 4;
      if (q.x > best || (q.x == best && (k + 0) < bcls)) { best = q.x; bcls = k + 0; }
      if (q.y > best || (q.y == best && (k + 1) < bcls)) { best = q.y; bcls = k + 1; }
      if (q.z > best || (q.z == best && (k + 2) < bcls)) { best = q.z; bcls = k + 2; }
      if (q.w > best || (q.w == best && (k + 3) < bcls)) { best = q.w; bcls = k + 3; }
    }
  }
#endif

  if (a < NANCH) {
    const long long idx = rowbase + t;
    float4 r  = reinterpret_cast<const float4*>(reg)[idx];
    float4 an = reinterpret_cast<const float4*>(anc)[idx];
    float aw  = an.z - an.x, ah = an.w - an.y;
    float acx = an.x + 0.5f * aw, acy = an.y + 0.5f * ah;
    float pcx = (r.x * 0.1f) * aw + acx;
    float pcy = (r.y * 0.1f) * ah + acy;
    float pw  = expf(r.z * 0.2f) * aw;
    float ph  = expf(r.w * 0.2f) * ah;
    float x1 = truncf(pcx - 0.5f * pw);
    float y1 = truncf(pcy - 0.5f * ph);
    float x2 = truncf(pcx + 0.5f * pw);
    float y2 = truncf(pcy + 0.5f * ph);
    x1 = fmaxf(x1, 0.0f);    y1 = fmaxf(y1, 0.0f);
    x2 = fminf(x2, IMG_WM1); y2 = fminf(y2, IMG_HM1);
    reinterpret_cast<float4*>(boxes)[idx] = make_float4(x1, y1, x2, y2);
    areas[idx]   = (x2 - x1) * (y2 - y1);
    scores[idx]  = (best > SCORE_THR) ? best : -__builtin_inff();
    classes[idx] = bcls;
  }
}

// ---------------------------------------------------------------------------
// Phase 2: greedy class-aware NMS, one workgroup per batch image.
// Live scores (300KB) held entirely in LDS (CDNA5: 320KB per WGP).
// ---------------------------------------------------------------------------
__global__ __launch_bounds__(P2_THREADS) void retina_nms_kernel(
    const float* __restrict__ boxes, const float* __restrict__ areas,
    const float* __restrict__ scores, const int* __restrict__ classes,
    float* __restrict__ out)
{
  extern __shared__ float s_scores[];              // NANCH floats (dynamic LDS)
  __shared__ float red_v[32];
  __shared__ int   red_i[32];
  __shared__ float bsel[8];  // score, x1,y1,x2,y2, area, class, ok

  const int b   = blockIdx.x;
  const int tid = threadIdx.x;
  const long long base = (long long)b * NANCH;
  const float4* bx = reinterpret_cast<const float4*>(boxes + base * 4);
  const float*  ar = areas + base;
  const int*    cl = classes + base;

  for (int i = tid; i < NANCH; i += P2_THREADS) s_scores[i] = scores[base + i];
  __syncthreads();

  float* out_s = out + (long long)b * MAX_DET;
  float* out_c = out + (long long)BATCH * MAX_DET + (long long)b * MAX_DET;
  float* out_b = out + 2ll * BATCH * MAX_DET + (long long)b * MAX_DET * 4;

  for (int det = 0; det < MAX_DET; ++det) {
    // ---- parallel argmax over LDS scores (first index wins on ties) ----
    float mv = -__builtin_inff();
    int   mi = 0x7FFFFFFF;
    for (int i = tid; i < NANCH; i += P2_THREADS) {
      float v = s_scores[i];                 // lane-consecutive: conflict-free
      if (v > mv) { mv = v; mi = i; }        // increasing i => lowest idx kept
    }
    for (int off = 16; off > 0; off >>= 1) { // wave32 reduce
      float ov = __shfl_down(mv, off);
      int   oi = __shfl_down(mi, off);
      if (ov > mv || (ov == mv && oi < mi)) { mv = ov; mi = oi; }
    }
    const int lane = tid & 31, wv = tid >> 5;
    if (lane == 0) { red_v[wv] = mv; red_i[wv] = mi; }
    __syncthreads();
    if (wv == 0) {
      mv = red_v[lane]; mi = red_i[lane];
      for (int off = 16; off > 0; off >>= 1) {
        float ov = __shfl_down(mv, off);
        int   oi = __shfl_down(mi, off);
        if (ov > mv || (ov == mv && oi < mi)) { mv = ov; mi = oi; }
      }
      if (lane == 0) {
        if (mv > -__builtin_inff()) {
          float4 sb = bx[mi];
          bsel[0] = mv;
          bsel[1] = sb.x; bsel[2] = sb.y; bsel[3] = sb.z; bsel[4] = sb.w;
          bsel[5] = ar[mi];
          bsel[6] = (float)cl[mi];
          bsel[7] = 1.0f;
          s_scores[mi] = -__builtin_inff();       // kill selected anchor
          out_s[det] = mv;
          out_c[det] = (float)cl[mi];
          out_b[det * 4 + 0] = sb.x; out_b[det * 4 + 1] = sb.y;
          out_b[det * 4 + 2] = sb.z; out_b[det * 4 + 3] = sb.w;
        } else {
          bsel[7] = 0.0f;
          out_s[det] = -1.0f;
          out_c[det] = -1.0f;
          out_b[det * 4 + 0] = -1.0f; out_b[det * 4 + 1] = -1.0f;
          out_b[det * 4 + 2] = -1.0f; out_b[det * 4 + 3] = -1.0f;
        }
      }
    }
    __syncthreads();

    // ---- suppression sweep (same-class IoU > 0.5) ----
    if (bsel[7] != 0.0f) {
      const float sx1 = bsel[1], sy1 = bsel[2], sx2 = bsel[3], sy2 = bsel[4];
      const float sarea = bsel[5];
      const int   scls  = (int)bsel[6];
      for (int i = tid; i < NANCH; i += P2_THREADS) {
        if (s_scores[i] == -__builtin_inff()) continue;
        if (cl[i] != scls) continue;
        float4 q = bx[i];
        float xx1 = fmaxf(q.x, sx1), yy1 = fmaxf(q.y, sy1);
        float xx2 = fminf(q.z, sx2), yy2 = fminf(q.w, sy2);
        float inter = fmaxf(xx2 - xx1, 0.0f) * fmaxf(yy2 - yy1, 0.0f);
        float uni   = ar[i] + sarea - inter;
        float iou   = (uni > 0.0f) ? inter / uni : 0.0f;
        if (iou > IOU_THR) s_scores[i] = -__builtin_inff();
      }
    }
    __syncthreads();
  }
}

// ---------------------------------------------------------------------------
extern "C" void kernel_launch(void* const* d_in, const int* in_sizes, int n_in,
                              void* d_out, int out_size, void* d_ws, size_t ws_size,
                              hipStream_t stream) {
  const float* cls = (const float*)d_in[0];   // [B, A, C]
  const float* reg = (const float*)d_in[1];   // [B, A, 4]
  const float* anc = (const float*)d_in[2];   // [B, A, 4]
  float* out = (float*)d_out;                 // ss[B*50] | cc[B*50] | bb[B*50*4]

  // Workspace layout (floats): boxes[B*A*4] | areas[B*A] | scores[B*A] | classes[B*A]
  float* boxes   = (float*)d_ws;
  float* areas   = boxes + (size_t)BATCH * NANCH * 4;
  float* scoresW = areas + (size_t)BATCH * NANCH;
  int*   classesW = (int*)(scoresW + (size_t)BATCH * NANCH);

  dim3 g1((NANCH + TILE - 1) / TILE, BATCH);
  retina_decode_kernel<<<g1, TILE, 0, stream>>>(cls, reg, anc,
                                                boxes, areas, scoresW, classesW);

  // 300KB dynamic LDS: entire live-score vector per batch image in one WGP.
  retina_nms_kernel<<<dim3(BATCH), P2_THREADS,
                      (size_t)NANCH * sizeof(float), stream>>>(
      boxes, areas, scoresW, classesW, out);
}